// YOLOv5MaxProbExtractor_82523501626081
// MI455X (gfx1250) — compile-verified
//
#include <hip/hip_runtime.h>
#include <cstdint>

#define NB0 19200   // 3*80*80
#define NB1 4800    // 3*40*40
#define NB2 1200    // 3*20*20
#define NBOX (NB0 + NB1 + NB2)  // 25200 boxes per batch
#define NBATCH 32

__device__ __forceinline__ float sigf(float v) {
    // sigmoid via hardware v_exp_f32 (memory-bound kernel; fast path is fine)
    return 1.0f / (1.0f + __expf(-v));
}

__global__ __launch_bounds__(64) void yolo_init(float* __restrict__ out) {
    int t = threadIdx.x;
    if (t < 33) out[t] = 0.0f;
}

__global__ __launch_bounds__(256) void yolo_main(const float* __restrict__ p0,
                                                 const float* __restrict__ p1,
                                                 const float* __restrict__ p2,
                                                 const float* __restrict__ gt,
                                                 const float* __restrict__ thr,
                                                 float* __restrict__ out) {
    const int b   = blockIdx.y;   // one batch per blockIdx.y -> clean per-batch reduction
    const int tid = threadIdx.x;

    __shared__ float s_params[8];   // gx1, gy1, gx2, gy2, thresh
    __shared__ float red[256];

    // --- CDNA5 async path: broadcast uniform params into LDS via the async
    // global->LDS DMA (tracked by ASYNCcnt, not LOADcnt). Lanes 0-3 fetch the
    // gt box, lane 4 fetches iou_thresh. ---
    if (tid < 5) {
        const float* src = (tid < 4) ? (gt + b * 4 + tid) : thr;
        unsigned lds_off = (unsigned)(uintptr_t)(&s_params[tid]);
        unsigned long long ga = (unsigned long long)(uintptr_t)src;
        asm volatile("global_load_async_to_lds_b32 %0, %1, off"
                     :: "v"(lds_off), "v"(ga) : "memory");
    }
    asm volatile("s_wait_asynccnt 0" ::: "memory");
    __syncthreads();

    const float gx1 = s_params[0], gy1 = s_params[1];
    const float gx2 = s_params[2], gy2 = s_params[3];
    const float thresh = s_params[4];
    const float area_g = (gx2 - gx1) * (gy2 - gy1);

    float best = 0.0f;

    for (int i = blockIdx.x * blockDim.x + tid; i < NBOX; i += gridDim.x * blockDim.x) {
        const float* rec;
        float stride, aw, ah;
        int xg, yg;
        if (i < NB0) {
            int j = i; int a = j / 6400; int r = j - a * 6400;
            yg = r / 80; xg = r - yg * 80;
            rec = p0 + ((size_t)b * NB0 + (size_t)j) * 85;
            stride = 8.0f;
            aw = (a == 0) ? 10.f  : (a == 1) ? 16.f  : 33.f;
            ah = (a == 0) ? 13.f  : (a == 1) ? 30.f  : 23.f;
        } else if (i < NB0 + NB1) {
            int j = i - NB0; int a = j / 1600; int r = j - a * 1600;
            yg = r / 40; xg = r - yg * 40;
            rec = p1 + ((size_t)b * NB1 + (size_t)j) * 85;
            stride = 16.0f;
            aw = (a == 0) ? 30.f  : (a == 1) ? 62.f  : 59.f;
            ah = (a == 0) ? 61.f  : (a == 1) ? 45.f  : 119.f;
        } else {
            int j = i - NB0 - NB1; int a = j / 400; int r = j - a * 400;
            yg = r / 20; xg = r - yg * 20;
            rec = p2 + ((size_t)b * NB2 + (size_t)j) * 85;
            stride = 32.0f;
            aw = (a == 0) ? 116.f : (a == 1) ? 156.f : 373.f;
            ah = (a == 0) ? 90.f  : (a == 1) ? 198.f : 326.f;
        }

        // Only channels 0..5 of 85 are needed: 24B of each 340B record.
        // Records are 4B-aligned only, so dword loads; NT hint (stream-once).
        float t0 = __builtin_nontemporal_load(rec + 0);
        float t1 = __builtin_nontemporal_load(rec + 1);
        float t2 = __builtin_nontemporal_load(rec + 2);
        float t3 = __builtin_nontemporal_load(rec + 3);
        float t4 = __builtin_nontemporal_load(rec + 4);
        float t5 = __builtin_nontemporal_load(rec + 5);

        float sx = sigf(t0), sy = sigf(t1), sw = sigf(t2), sh = sigf(t3);
        float cx = (sx * 2.0f - 0.5f + (float)xg) * stride;
        float cy = (sy * 2.0f - 0.5f + (float)yg) * stride;
        float w  = (sw * 2.0f) * (sw * 2.0f) * aw;
        float h  = (sh * 2.0f) * (sh * 2.0f) * ah;
        float bx1 = cx - 0.5f * w, bx2 = cx + 0.5f * w;
        float by1 = cy - 0.5f * h, by2 = cy + 0.5f * h;

        float iw = fmaxf(fminf(bx2, gx2) - fmaxf(bx1, gx1), 0.0f);
        float ih = fmaxf(fminf(by2, gy2) - fmaxf(by1, gy1), 0.0f);
        float inter  = iw * ih;
        float area_b = w * h;   // (bx2-bx1)*(by2-by1) == w*h exactly
        float iou    = inter / (area_b + area_g - inter);

        float score = sigf(t4) * sigf(t5);
        best = fmaxf(best, (iou >= thresh) ? score : 0.0f);
    }

    // Block max-reduction, then one atomic per block. Scores are >= 0, so
    // integer atomicMax on the IEEE bit pattern is order-exact.
    red[tid] = best;
    __syncthreads();
    #pragma unroll
    for (int s = 128; s > 0; s >>= 1) {
        if (tid < s) red[tid] = fmaxf(red[tid], red[tid + s]);
        __syncthreads();
    }
    if (tid == 0) {
        atomicMax(reinterpret_cast<unsigned*>(out + 1 + b), __float_as_uint(red[0]));
    }
}

__global__ __launch_bounds__(32) void yolo_finalize(float* __restrict__ out) {
    __shared__ float s[32];
    int t = threadIdx.x;
    s[t] = out[1 + t];
    __syncthreads();
    #pragma unroll
    for (int k = 16; k > 0; k >>= 1) {
        if (t < k) s[t] += s[t + k];
        __syncthreads();
    }
    if (t == 0) out[0] = s[0] * (1.0f / 32.0f);
}

extern "C" void kernel_launch(void* const* d_in, const int* in_sizes, int n_in,
                              void* d_out, int out_size, void* d_ws, size_t ws_size,
                              hipStream_t stream) {
    (void)in_sizes; (void)n_in; (void)out_size; (void)d_ws; (void)ws_size;
    const float* p0  = (const float*)d_in[0];
    const float* p1  = (const float*)d_in[1];
    const float* p2  = (const float*)d_in[2];
    const float* gt  = (const float*)d_in[3];
    const float* thr = (const float*)d_in[4];
    float* out = (float*)d_out;

    yolo_init<<<1, 64, 0, stream>>>(out);

    dim3 grid(40, NBATCH);   // 40 blocks strided over 25200 boxes of one batch
    yolo_main<<<grid, 256, 0, stream>>>(p0, p1, p2, gt, thr, out);

    yolo_finalize<<<1, 32, 0, stream>>>(out);
}